// GraphSAGE_40510131535944
// MI455X (gfx1250) — compile-verified
//
#include <hip/hip_runtime.h>
#include <hip/hip_bf16.h>

typedef __attribute__((ext_vector_type(2))) float v2f;
typedef __attribute__((ext_vector_type(8))) float v8f;

#define NNODES 100000
#define DIM    128
#define NEDGES 1600000

// ---------------------------------------------------------------------------
// deg[i] = 1.0 (self loop)
__global__ void sage_init_deg(float* __restrict__ deg) {
    int i = blockIdx.x * blockDim.x + threadIdx.x;
    if (i < NNODES) deg[i] = 1.0f;
}

// deg[dst[e]] += 1 ; deg[src[e]] += 1   (bidirected graph in-degree)
__global__ void sage_degree(float* __restrict__ deg,
                            const int* __restrict__ src,
                            const int* __restrict__ dst) {
    int e = blockIdx.x * blockDim.x + threadIdx.x;
    if (e >= NEDGES) return;
    atomicAdd(&deg[dst[e]], 1.0f);
    atomicAdd(&deg[src[e]], 1.0f);
}

// neigh = h  (self-loop message), float4 vectorized
__global__ void sage_copy(float* __restrict__ out, const float* __restrict__ in) {
    long long i = (long long)blockIdx.x * blockDim.x + threadIdx.x;
    long long n4 = (long long)NNODES * DIM / 4;
    if (i >= n4) return;
    ((float4*)out)[i] = ((const float4*)in)[i];
}

// scatter-add: message u->v, 32 lanes per message, float4 per lane (128 feats)
__global__ void sage_scatter(const float* __restrict__ hin,
                             float* __restrict__ neigh,
                             const int* __restrict__ src,
                             const int* __restrict__ dst) {
    long long t = (long long)blockIdx.x * blockDim.x + threadIdx.x;
    const long long total = (long long)NEDGES * 2 * 32;
    if (t >= total) return;
    int lane = (int)(t & 31);
    long long msg = t >> 5;
    int u, v;
    if (msg < NEDGES) { u = src[msg];            v = dst[msg]; }
    else              { u = dst[msg - NEDGES];   v = src[msg - NEDGES]; }
    const float4 x = *(const float4*)&hin[(size_t)u * DIM + lane * 4];
    float* p = &neigh[(size_t)v * DIM + lane * 4];
    atomicAdd(p + 0, x.x);
    atomicAdd(p + 1, x.y);
    atomicAdd(p + 2, x.z);
    atomicAdd(p + 3, x.w);
}

// hout = hin @ Wself^T + bias + (neigh * inv_deg) @ Wneigh^T
// Block = 256 threads = 8 waves; block owns 16 rows; wave w owns cols [16w,16w+16).
// Per wave: 32 k-steps (K=4) x 2 matrices = 64 v_wmma_f32_16x16x4_f32.
__global__ void __launch_bounds__(256)
sage_gemm(const float* __restrict__ hin,
          const float* __restrict__ neigh,
          const float* __restrict__ Wself,   // [DIM, DIM] row-major (out-feature major)
          const float* __restrict__ bias,    // [DIM]
          const float* __restrict__ Wneigh,  // [DIM, DIM]
          const float* __restrict__ deg,     // [NNODES]
          float* __restrict__ hout) {
    const int wave = threadIdx.x >> 5;   // 0..7  (wave32)
    const int lane = threadIdx.x & 31;
    const int m    = lane & 15;
    const int half = lane >> 4;          // 0 or 1
    const int node = blockIdx.x * 16;
    const int row  = node + m;           // A-fragment row for this lane
    const int ncol = wave * 16 + m;      // B/C/D column for this lane

    const float b = bias[ncol];
    v8f acc = { b, b, b, b, b, b, b, b };

    const float idg = 1.0f / deg[row];

    const float* hrow = hin    + (size_t)row  * DIM;
    const float* nrow = neigh  + (size_t)row  * DIM;
    const float* wsr  = Wself  + (size_t)ncol * DIM;
    const float* wnr  = Wneigh + (size_t)ncol * DIM;

#pragma unroll
    for (int k0 = 0; k0 < DIM; k0 += 4) {
        const int kc = k0 + 2 * half;    // A VGPR0: K={0,2}, VGPR1: K={1,3}
        v2f a_s = *(const v2f*)(hrow + kc);
        v2f a_n = *(const v2f*)(nrow + kc);
        a_n *= idg;                      // fold mean normalization into A frag
        v2f b_s = *(const v2f*)(wsr + kc);   // B[k,n] = W[n,k]
        v2f b_n = *(const v2f*)(wnr + kc);
        acc = __builtin_amdgcn_wmma_f32_16x16x4_f32(
                  false, a_s, false, b_s, (short)0, acc, false, false);
        acc = __builtin_amdgcn_wmma_f32_16x16x4_f32(
                  false, a_n, false, b_n, (short)0, acc, false, false);
    }

    // C/D layout: VGPR r -> row = r + 8*half, col = lane&15
    float* orow = hout + (size_t)(node + 8 * half) * DIM + wave * 16 + m;
#pragma unroll
    for (int r = 0; r < 8; ++r)
        orow[(size_t)r * DIM] = acc[r];
}

// ---------------------------------------------------------------------------
extern "C" void kernel_launch(void* const* d_in, const int* in_sizes, int n_in,
                              void* d_out, int out_size, void* d_ws, size_t ws_size,
                              hipStream_t stream) {
    const float* h      = (const float*)d_in[0];  // [N, D]
    const float* Wself  = (const float*)d_in[1];  // [L, D, D]
    const float* bself  = (const float*)d_in[2];  // [L, D]
    const float* Wneigh = (const float*)d_in[3];  // [L, D, D]
    const int*   src    = (const int*)d_in[4];    // [E]
    const int*   dst    = (const int*)d_in[5];    // [E]
    float*       out    = (float*)d_out;          // [N, D]

    char* ws = (char*)d_ws;
    float* deg  = (float*)ws;                                            // N floats
    float* hbuf = (float*)(ws + (size_t)(1 << 20));                      // N*D floats
    float* nbuf = (float*)(ws + (size_t)(1 << 20) +
                           (size_t)NNODES * DIM * sizeof(float));        // N*D floats

    const int T = 256;
    const long long copyN    = (long long)NNODES * DIM / 4;
    const long long scatterN = (long long)NEDGES * 2 * 32;
    const int copyB    = (int)((copyN    + T - 1) / T);
    const int scatterB = (int)((scatterN + T - 1) / T);
    const int gemmB    = NNODES / 16;  // 6250, exact

    // degree (graph-structural, computed once per launch)
    sage_init_deg<<<(NNODES + T - 1) / T, T, 0, stream>>>(deg);
    sage_degree<<<(NEDGES + T - 1) / T, T, 0, stream>>>(deg, src, dst);

    // ---- layer 0: h (input) -> hbuf
    sage_copy<<<copyB, T, 0, stream>>>(nbuf, h);
    sage_scatter<<<scatterB, T, 0, stream>>>(h, nbuf, src, dst);
    sage_gemm<<<gemmB, T, 0, stream>>>(h, nbuf,
                                       Wself, bself, Wneigh, deg, hbuf);

    // ---- layer 1: hbuf -> out
    sage_copy<<<copyB, T, 0, stream>>>(nbuf, hbuf);
    sage_scatter<<<scatterB, T, 0, stream>>>(hbuf, nbuf, src, dst);
    sage_gemm<<<gemmB, T, 0, stream>>>(hbuf, nbuf,
                                       Wself + DIM * DIM, bself + DIM,
                                       Wneigh + DIM * DIM, deg, out);
}